// Pool_15135464751208
// MI455X (gfx1250) — compile-verified
//
#include <hip/hip_runtime.h>
#include <math.h>

#define B  4
#define N  6144
#define D  256
#define KK 3072

typedef __attribute__((ext_vector_type(2))) float v2f;
typedef __attribute__((ext_vector_type(8))) float v8f;
typedef __attribute__((ext_vector_type(4))) int   v4i;

#if __has_builtin(__builtin_amdgcn_global_load_async_to_lds_b128) && \
    __has_builtin(__builtin_amdgcn_s_wait_asynccnt)
#define USE_ASYNC_LDS 1
#else
#define USE_ASYNC_LDS 0
#endif

#define AS_LDS    __attribute__((address_space(3)))
#define AS_GLOBAL __attribute__((address_space(1)))

// ---------------------------------------------------------------------------
// Kernel S: scores[b][n] = sigmoid(dot(h[b,n,:], W) + bias)
// One wave32 computes a 16-row tile via V_WMMA_F32_16X16X4_F32, K looped 4 at
// a time. W is broadcast to all 16 B-columns, so column 0 of D holds the
// mat-vec result (extracted from lanes 0 and 16 per the 16x16 f32 C/D layout).
// ---------------------------------------------------------------------------
__global__ __launch_bounds__(256) void score_kernel(const float* __restrict__ h,
                                                    const float* __restrict__ W,
                                                    const float* __restrict__ bias_p,
                                                    float* __restrict__ scores) {
  const int lane = threadIdx.x & 31;
  const int wave = blockIdx.x * 8 + (threadIdx.x >> 5);
  const int b    = wave / (N / 16);
  const int row0 = (wave % (N / 16)) * 16;
  const float bias = bias_p[0];

  const int m    = lane & 15;
  const int koff = (lane >> 4) << 1;  // lanes 0-15: K+0/K+1, lanes 16-31: K+2/K+3
  const float* hrow = h + ((size_t)b * N + row0 + m) * D;

#if __has_builtin(__builtin_amdgcn_wmma_f32_16x16x4_f32)
  v8f acc = {};
  for (int k = 0; k < D; k += 4) {
    v2f a  = { hrow[k + koff], hrow[k + koff + 1] };
    v2f bm = { W[k + koff],    W[k + koff + 1]    };  // W broadcast across N
    acc = __builtin_amdgcn_wmma_f32_16x16x4_f32(false, a, false, bm,
                                                (short)0, acc, false, false);
  }
  // Column n=0 lives in lane 0 (M=0..7 in acc[0..7]) and lane 16 (M=8..15).
  if (lane == 0 || lane == 16) {
    const int mb = (lane == 0) ? 0 : 8;
    for (int r = 0; r < 8; ++r) {
      float x = acc[r] + bias;
      scores[(size_t)b * N + row0 + mb + r] = 1.0f / (1.0f + expf(-x));
    }
  }
#else
  if (lane < 16) {
    float s = 0.0f;
    for (int k = 0; k < D; ++k) s += hrow[k] * W[k];
    float x = s + bias;
    scores[(size_t)b * N + row0 + m] = 1.0f / (1.0f + expf(-x));
  }
#endif
}

// ---------------------------------------------------------------------------
// Kernel T: exact jax.lax.top_k via ranking. rank(i) = #{j: s_j>s_i or
// (s_j==s_i and j<i)} is a bijection, so scattering i -> rank yields values
// sorted descending with stable index tie-break. 8 blocks per batch; scores
// staged in LDS, inner loop is an LDS broadcast read.
// ---------------------------------------------------------------------------
__global__ __launch_bounds__(768) void topk_kernel(const float* __restrict__ scores,
                                                   int* __restrict__ widx,
                                                   float* __restrict__ wval) {
  __shared__ float s[N];
  const int b    = blockIdx.x >> 3;
  const int part = blockIdx.x & 7;
  for (int j = threadIdx.x; j < N; j += 768) s[j] = scores[(size_t)b * N + j];
  __syncthreads();

  const int   i  = part * 768 + threadIdx.x;
  const float si = s[i];
  int rank = 0;
  for (int j = 0; j < N; ++j) {
    const float sj = s[j];
    rank += ((sj > si) || ((sj == si) && (j < i))) ? 1 : 0;
  }
  if (rank < KK) {
    widx[b * KK + rank] = i;
    wval[b * KK + rank] = si;
  }
}

// ---------------------------------------------------------------------------
// Kernel H: new_h[b][i][:] = h[b][idx[b][i]][:] * values[b][i]   (coalesced)
// ---------------------------------------------------------------------------
__global__ __launch_bounds__(256) void newh_kernel(const float* __restrict__ h,
                                                   const int* __restrict__ widx,
                                                   const float* __restrict__ wval,
                                                   float* __restrict__ out) {
  const int bi = blockIdx.x;           // 0 .. B*KK-1
  const int b  = bi / KK, i = bi % KK;
  const int   r = widx[b * KK + i];
  const float v = wval[b * KK + i];
  out[(size_t)bi * D + threadIdx.x] =
      h[((size_t)b * N + r) * D + threadIdx.x] * v;
}

// ---------------------------------------------------------------------------
// Kernel G: row/column gather. One block per output row: stage the full 24KB
// source row into LDS via async global->LDS B128 copies (VGPR bypass, tracked
// by ASYNCcnt), gather the 3072 selected columns from LDS, store coalesced.
// Optionally dual-writes (g_sentence[2:] == g_section[2:]). Prefetches the
// next selected row into L2 (global_prefetch_b8).
// ---------------------------------------------------------------------------
__global__ __launch_bounds__(256) void gather_kernel(const float* __restrict__ g,
                                                     const int* __restrict__ widx,
                                                     float* __restrict__ out0,
                                                     float* __restrict__ out1,
                                                     int dupFromB) {
  __shared__ float row[N];
  __shared__ int   cols[KK];
  const int i   = blockIdx.x;
  const int b   = blockIdx.y;
  const int tid = threadIdx.x;

  const int r = widx[b * KK + i];
  const float* srcf = g + ((size_t)b * N + r) * N;

#if USE_ASYNC_LDS
  {
    AS_GLOBAL v4i* gsrc = (AS_GLOBAL v4i*)srcf;   // addrspacecast + const drop
    AS_LDS v4i*    lrow = (AS_LDS v4i*)row;       // generic -> LDS
    // 1536 x 16B chunks / 256 threads = 6 async B128 copies per thread
    for (int j = tid; j < N / 4; j += 256) {
      __builtin_amdgcn_global_load_async_to_lds_b128(gsrc + j, lrow + j, 0, 0);
    }
  }
#else
  {
    const float4* src = (const float4*)srcf;
    for (int j = tid; j < N / 4; j += 256) ((float4*)row)[j] = src[j];
  }
#endif

  for (int j = tid; j < KK; j += 256) cols[j] = widx[b * KK + j];

  if (i + 1 < KK) {  // warm L2 for the next selected row (scattered row order)
    const int rn = widx[b * KK + i + 1];
    __builtin_prefetch(g + ((size_t)b * N + rn) * N + (size_t)tid * 24, 0, 0);
  }

#if USE_ASYNC_LDS
  __builtin_amdgcn_s_wait_asynccnt(0);  // this wave's async LDS writes done
#endif
  __syncthreads();                      // all waves' writes visible

  float* dst0 = out0 + ((size_t)b * KK + i) * KK;
  const bool dup = (out1 != nullptr) && (b >= dupFromB);
  if (dup) {
    float* dst1 = out1 + ((size_t)b * KK + i) * KK;
    for (int j = tid; j < KK; j += 256) {
      const float v = row[cols[j]];
      dst0[j] = v;
      dst1[j] = v;
    }
  } else {
    for (int j = tid; j < KK; j += 256) dst0[j] = row[cols[j]];
  }
}

// ---------------------------------------------------------------------------
extern "C" void kernel_launch(void* const* d_in, const int* in_sizes, int n_in,
                              void* d_out, int out_size, void* d_ws, size_t ws_size,
                              hipStream_t stream) {
  (void)in_sizes; (void)n_in; (void)out_size; (void)ws_size;

  const float* g1 = (const float*)d_in[0];
  const float* g2 = (const float*)d_in[1];
  const float* h  = (const float*)d_in[2];
  const float* W  = (const float*)d_in[3];
  const float* bb = (const float*)d_in[4];
  // d_in[5] = k (== 3072, baked in as KK)

  // workspace layout: scores[B*N] f32 | widx[B*KK] i32 | wval[B*KK] f32
  float* scores = (float*)d_ws;
  int*   widx   = (int*)((char*)d_ws + (size_t)B * N * sizeof(float));
  float* wval   = (float*)((char*)d_ws + (size_t)B * N * sizeof(float)
                                       + (size_t)B * KK * sizeof(int));

  float* out        = (float*)d_out;
  float* g_section  = out;                              // [B,KK,KK]
  float* g_sentence = out + (size_t)B * KK * KK;        // [B,KK,KK]
  float* new_h      = out + 2 * (size_t)B * KK * KK;    // [B,KK,D]

  // scores: B*N/16 = 1536 waves, 8 waves per block -> 192 blocks
  score_kernel<<<(B * N / 16) / 8, 256, 0, stream>>>(h, W, bb, scores);

  // exact top-k by ranking: 8 blocks/batch * 768 threads = 6144 elems/batch
  topk_kernel<<<B * 8, 768, 0, stream>>>(scores, widx, wval);

  // gated feature gather
  newh_kernel<<<B * KK, D, 0, stream>>>(h, widx, wval, new_h);

  // g_section from g1 for all batches; batches >= 2 dual-write g_sentence
  gather_kernel<<<dim3(KK, B), 256, 0, stream>>>(g1, widx, g_section, g_sentence, 2);

  // g_sentence batches 0,1 from g2
  gather_kernel<<<dim3(KK, 2), 256, 0, stream>>>(g2, widx, g_sentence, nullptr, 0);
}